// SmoothedCrossEntropyLoss_1425929142510
// MI455X (gfx1250) — compile-verified
//
#include <hip/hip_runtime.h>
#include <hip/hip_bf16.h>
#include <math.h>

// CDNA5 / gfx1250, wave32. One block (8 waves) per row of V=32000 f32.
// Guard-free streaming main loop: clause of b128 non-temporal loads (512 MB
// read once > 192 MB L2, so NT). Both the finite-masked value sums AND the
// finite counts are folded through V_WMMA_F32_16X16X4_F32 (C += A x ones):
// per component only cmp_class + 2 cndmask remain in the VALU stream.
// Count in f32 is exact (integers < 2^24). Bandwidth-bound: ~22 us floor.

typedef __attribute__((ext_vector_type(2))) float v2f;
typedef __attribute__((ext_vector_type(4))) float v4f;
typedef __attribute__((ext_vector_type(8))) float v8f;

#define WAVES_PER_BLOCK 8
#define BLOCK (WAVES_PER_BLOCK * 32)

__device__ __forceinline__ bool is_finite_f32(float x) {
    // false for NaN and +-inf
    return fabsf(x) < __builtin_inff();
}

__global__ __launch_bounds__(BLOCK) void sce_row_reduce(
    const float* __restrict__ logp,
    const long long* __restrict__ target,
    const float* __restrict__ smoothing_p,
    float* __restrict__ row_loss,
    float* __restrict__ row_valid,
    int V)
{
    const int row  = blockIdx.x;
    const int tid  = threadIdx.x;
    const int lane = tid & 31;
    const int wave = tid >> 5;

    const float* lp  = logp + (size_t)row * (size_t)V;
    const v4f*   lp4 = (const v4f*)lp;

    const int Vvec      = V >> 2;             // # of float4 chunks
    const int fullIters = Vvec / BLOCK;       // guard-free iterations

    v8f c  = {};                              // WMMA accumulator: value sums
    v8f cc = {};                              // WMMA accumulator: finite counts
    v2f bones; bones.x = 1.0f; bones.y = 1.0f;

    // ---- main loop: no predication, pure streaming ----
    for (int it = 0; it < fullIters; ++it) {
        int idx = it * BLOCK + tid;
        v4f v = __builtin_nontemporal_load(&lp4[idx]);
        bool f0 = is_finite_f32(v.x);
        bool f1 = is_finite_f32(v.y);
        bool f2 = is_finite_f32(v.z);
        bool f3 = is_finite_f32(v.w);
        v2f a0, a1, n0, n1;
        a0.x = f0 ? v.x : 0.0f;   n0.x = f0 ? 1.0f : 0.0f;
        a0.y = f1 ? v.y : 0.0f;   n0.y = f1 ? 1.0f : 0.0f;
        a1.x = f2 ? v.z : 0.0f;   n1.x = f2 ? 1.0f : 0.0f;
        a1.y = f3 ? v.w : 0.0f;   n1.y = f3 ? 1.0f : 0.0f;
        c  = __builtin_amdgcn_wmma_f32_16x16x4_f32(false, a0, false, bones,
                                                   (short)0, c,  false, false);
        c  = __builtin_amdgcn_wmma_f32_16x16x4_f32(false, a1, false, bones,
                                                   (short)0, c,  false, false);
        cc = __builtin_amdgcn_wmma_f32_16x16x4_f32(false, n0, false, bones,
                                                   (short)0, cc, false, false);
        cc = __builtin_amdgcn_wmma_f32_16x16x4_f32(false, n1, false, bones,
                                                   (short)0, cc, false, false);
    }

    // ---- vec tail (block-uniform branch; EXEC all-ones at the WMMAs) ----
    if (fullIters * BLOCK < Vvec) {
        int idx = fullIters * BLOCK + tid;
        bool inr = idx < Vvec;
        v4f v = {};
        if (inr) v = __builtin_nontemporal_load(&lp4[idx]);
        bool f0 = inr && is_finite_f32(v.x);
        bool f1 = inr && is_finite_f32(v.y);
        bool f2 = inr && is_finite_f32(v.z);
        bool f3 = inr && is_finite_f32(v.w);
        v2f a0, a1, n0, n1;
        a0.x = f0 ? v.x : 0.0f;   n0.x = f0 ? 1.0f : 0.0f;
        a0.y = f1 ? v.y : 0.0f;   n0.y = f1 ? 1.0f : 0.0f;
        a1.x = f2 ? v.z : 0.0f;   n1.x = f2 ? 1.0f : 0.0f;
        a1.y = f3 ? v.w : 0.0f;   n1.y = f3 ? 1.0f : 0.0f;
        c  = __builtin_amdgcn_wmma_f32_16x16x4_f32(false, a0, false, bones,
                                                   (short)0, c,  false, false);
        c  = __builtin_amdgcn_wmma_f32_16x16x4_f32(false, a1, false, bones,
                                                   (short)0, c,  false, false);
        cc = __builtin_amdgcn_wmma_f32_16x16x4_f32(false, n0, false, bones,
                                                   (short)0, cc, false, false);
        cc = __builtin_amdgcn_wmma_f32_16x16x4_f32(false, n1, false, bones,
                                                   (short)0, cc, false, false);
    }

    // ---- scalar tail if V not a multiple of 4 (empty for V=32000) ----
    float s_extra = 0.0f;
    float n_extra = 0.0f;
    for (int idx = (Vvec << 2) + tid; idx < V; idx += BLOCK) {
        float x = lp[idx];
        bool f = is_finite_f32(x);
        s_extra += f ? x : 0.0f;
        n_extra += f ? 1.0f : 0.0f;
    }

    // Each accumulator holds the 16 row-sums replicated 16x (once per N col):
    // full-wave sum of per-lane partials = 16 * (wave total). /16 is exact.
    float c_lane  = c[0]  + c[1]  + c[2]  + c[3]  + c[4]  + c[5]  + c[6]  + c[7];
    float cc_lane = cc[0] + cc[1] + cc[2] + cc[3] + cc[4] + cc[5] + cc[6] + cc[7];
    for (int m = 16; m >= 1; m >>= 1) {
        c_lane  += __shfl_xor(c_lane,  m, 32);
        cc_lane += __shfl_xor(cc_lane, m, 32);
        s_extra += __shfl_xor(s_extra, m, 32);
        n_extra += __shfl_xor(n_extra, m, 32);
    }
    float wave_sum = c_lane  * (1.0f / 16.0f) + s_extra;
    float wave_cnt = cc_lane * (1.0f / 16.0f) + n_extra;   // exact integer in f32

    __shared__ float ls[WAVES_PER_BLOCK];
    __shared__ float lc[WAVES_PER_BLOCK];
    if (lane == 0) { ls[wave] = wave_sum; lc[wave] = wave_cnt; }
    __syncthreads();

    if (tid == 0) {
        float S = 0.0f, nf = 0.0f;
        for (int w = 0; w < WAVES_PER_BLOCK; ++w) { S += ls[w]; nf += lc[w]; }
        long long tgt = target[row];
        bool valid = (tgt != -1LL);
        float t = 0.0f;
        if (valid) {
            float x = lp[tgt];
            t = is_finite_f32(x) ? x : 0.0f;
        }
        float sm   = *smoothing_p;
        float prob = sm / fmaxf(nf, 1.0f);
        float loss = valid ? -(prob * (S - t) + (1.0f - sm) * t) : 0.0f;
        row_loss[row]  = loss;
        row_valid[row] = valid ? 1.0f : 0.0f;
    }
}

__global__ __launch_bounds__(256) void sce_finalize(
    const float* __restrict__ row_loss,
    const float* __restrict__ row_valid,
    float* __restrict__ out, int n)
{
    __shared__ float sl[256];
    __shared__ float sv[256];
    float L = 0.0f, C = 0.0f;
    for (int i = threadIdx.x; i < n; i += 256) {
        L += row_loss[i];
        C += row_valid[i];
    }
    sl[threadIdx.x] = L; sv[threadIdx.x] = C;
    __syncthreads();
    for (int off = 128; off >= 1; off >>= 1) {
        if ((int)threadIdx.x < off) {
            sl[threadIdx.x] += sl[threadIdx.x + off];
            sv[threadIdx.x] += sv[threadIdx.x + off];
        }
        __syncthreads();
    }
    if (threadIdx.x == 0) {
        float total = sl[0], count = sv[0];
        out[0] = (count > 0.0f) ? total / fmaxf(count, 1.0f) : 0.0f;
    }
}

extern "C" void kernel_launch(void* const* d_in, const int* in_sizes, int n_in,
                              void* d_out, int out_size, void* d_ws, size_t ws_size,
                              hipStream_t stream) {
    const float*     logp      = (const float*)d_in[0];
    const long long* target    = (const long long*)d_in[1];   // int64 targets
    const float*     smoothing = (const float*)d_in[2];

    const int N = in_sizes[1];            // B*T rows
    const int V = in_sizes[0] / N;        // vocab

    float* row_loss  = (float*)d_ws;      // N floats (rewritten every call)
    float* row_valid = row_loss + N;      // N floats (rewritten every call)

    sce_row_reduce<<<N, BLOCK, 0, stream>>>(logp, target, smoothing,
                                            row_loss, row_valid, V);
    sce_finalize<<<1, 256, 0, stream>>>(row_loss, row_valid, (float*)d_out, N);
}